// PointNetSetAbstraction_2534030705297
// MI455X (gfx1250) — compile-verified
//
#include <hip/hip_runtime.h>
#include <hip/hip_bf16.h>

typedef __attribute__((ext_vector_type(2))) float v2f;
typedef __attribute__((ext_vector_type(8))) float v8f;
typedef __attribute__((ext_vector_type(4))) unsigned int u32x4;
typedef __attribute__((ext_vector_type(8))) int i32x8;
typedef __attribute__((ext_vector_type(4))) int i32x4;

#define BATCH 16
#define NPTS  4096
#define S_OUT 1024
#define KSAMP 32
#define M_ROWS ((size_t)BATCH * S_OUT * KSAMP)   // 524288
#define R2 (0.2f * 0.2f)

// ---------------------------------------------------------------------------
// 1) Farthest point sampling: one block per batch, dist[] lives in LDS.
// ---------------------------------------------------------------------------
__global__ void __launch_bounds__(256) fps_kernel(const float* __restrict__ xyz,
                                                  float* __restrict__ out_newxyz) {
  __shared__ float dist[NPTS];
  __shared__ float rval[256];
  __shared__ int   ridx[256];
  __shared__ int   s_far;
  const int b = blockIdx.x, tid = threadIdx.x;
  const float* X = xyz + (size_t)b * NPTS * 3;
  for (int i = tid; i < NPTS; i += 256) dist[i] = 1e10f;
  if (tid == 0) s_far = 0;
  __syncthreads();
  for (int step = 0; step < S_OUT; ++step) {
    const int far = s_far;
    const float cx = X[far * 3 + 0], cy = X[far * 3 + 1], cz = X[far * 3 + 2];
    if (tid == 0) {
      float* o = out_newxyz + ((size_t)b * S_OUT + step) * 3;
      o[0] = cx; o[1] = cy; o[2] = cz;
    }
    float bestv = -1.0f; int besti = 0;
    for (int i = tid; i < NPTS; i += 256) {
      const float dx = X[i * 3 + 0] - cx;
      const float dy = X[i * 3 + 1] - cy;
      const float dz = X[i * 3 + 2] - cz;
      const float d  = dx * dx + dy * dy + dz * dz;
      const float nd = fminf(dist[i], d);
      dist[i] = nd;
      if (nd > bestv) { bestv = nd; besti = i; }   // strictly-greater -> first max
    }
    rval[tid] = bestv; ridx[tid] = besti;
    __syncthreads();
    for (int s = 128; s > 0; s >>= 1) {
      if (tid < s) {
        const float v2 = rval[tid + s]; const int i2 = ridx[tid + s];
        if (v2 > rval[tid] || (v2 == rval[tid] && i2 < ridx[tid])) { rval[tid] = v2; ridx[tid] = i2; }
      }
      __syncthreads();
    }
    if (tid == 0) s_far = ridx[0];
    __syncthreads();
  }
}

// ---------------------------------------------------------------------------
// 2) Ball query: one wave32 per centroid; ballot + prefix-popcount appends the
//    first 32 in-radius indices in ascending order (== sort-then-slice).
// ---------------------------------------------------------------------------
__global__ void __launch_bounds__(256) ballq_kernel(const float* __restrict__ xyz,
                                                    const float* __restrict__ newxyz,
                                                    int* __restrict__ idxbuf) {
  const int gw   = (int)((blockIdx.x * blockDim.x + threadIdx.x) >> 5);
  const int lane = threadIdx.x & 31;
  const int b = gw >> 10;              // / S_OUT
  const float* X = xyz + (size_t)b * NPTS * 3;
  const float* c = newxyz + (size_t)gw * 3;
  const float cx = c[0], cy = c[1], cz = c[2];
  const float cn = cx * cx + cy * cy + cz * cz;
  int* out = idxbuf + (size_t)gw * KSAMP;

  int cnt = 0, first = 0;
  bool havefirst = false;
  for (int base = 0; base < NPTS && cnt < KSAMP; base += 32) {
    const int i = base + lane;
    const float px = X[i * 3 + 0], py = X[i * 3 + 1], pz = X[i * 3 + 2];
    const float pn  = px * px + py * py + pz * pz;
    const float dot = px * cx + py * cy + pz * cz;
    const float d   = cn + pn - 2.0f * dot;        // same formula as square_distance
    const bool inb  = (d <= R2);
    const unsigned mask = (unsigned)__ballot(inb); // wave32: low 32 bits
    if (mask) {
      if (!havefirst) { first = base + (__ffs(mask) - 1); havefirst = true; }
      const int prefix = __popc(mask & ((1u << lane) - 1u));
      if (inb && (cnt + prefix) < KSAMP) out[cnt + prefix] = i;
      cnt += __popc(mask);
    }
  }
  for (int j = cnt + lane; j < KSAMP; j += 32) out[j] = first;
}

// ---------------------------------------------------------------------------
// 3) Gather + recenter + concat -> X0[M][8]  (cols 0..2 rel xyz, 3..5 feats)
// ---------------------------------------------------------------------------
__global__ void __launch_bounds__(256) group_kernel(const float* __restrict__ xyz,
                                                    const float* __restrict__ pts,
                                                    const float* __restrict__ newxyz,
                                                    const int* __restrict__ idxbuf,
                                                    float* __restrict__ x0) {
  const size_t m = (size_t)blockIdx.x * blockDim.x + threadIdx.x;
  if (m >= M_ROWS) return;
  const size_t bs = m >> 5;            // (b,s) flat
  const int b = (int)(bs >> 10);
  const int id = idxbuf[m];
  const float* p = xyz + ((size_t)b * NPTS + id) * 3;
  const float* q = pts + ((size_t)b * NPTS + id) * 3;
  const float* c = newxyz + bs * 3;
  float* o = x0 + m * 8;
  o[0] = p[0] - c[0]; o[1] = p[1] - c[1]; o[2] = p[2] - c[2];
  o[3] = q[0];        o[4] = q[1];        o[5] = q[2];
  o[6] = 0.0f;        o[7] = 0.0f;
}

// ---------------------------------------------------------------------------
// TDM: load a dense 2D fp32 matrix (rows x cols, contiguous) into LDS.
// D# built per CDNA5 ISA 8.3/8.4: group0 = {count=1 | lds_addr | global_addr |
// type=2}, group1 = {data_size=4B, tensor_dim0=cols, tensor_dim1=rows,
// tile_dim0=cols, tile_dim1=rows, dim0_stride=cols}. Groups 2/3 unused (2D).
// This toolchain carries the 6-arg builtin (extra trailing i32x8 group state).
// Issued by one wave; completion via s_wait_tensorcnt.
// ---------------------------------------------------------------------------
__device__ __forceinline__ void tdm_load_2d(const float* w, float* ldsdst,
                                            int rows, int cols) {
  const unsigned lds_off = (unsigned)(size_t)ldsdst;        // low 32b of flat LDS addr
  const unsigned long long ga = (unsigned long long)(size_t)w;
  u32x4 g0;
  g0[0] = 1u;                                               // count=1, no gather
  g0[1] = lds_off;                                          // lds_addr
  g0[2] = (unsigned)(ga & 0xFFFFFFFFu);                     // global_addr[31:0]
  g0[3] = (unsigned)((ga >> 32) & 0x01FFFFFFu) | (2u << 30);// global_addr[56:32] | type=2
  i32x8 g1;
  g1[0] = (int)(2u << 16);                                  // data_size = 4 bytes
  g1[1] = (int)((unsigned)cols << 16);                      // tensor_dim0[15:0]
  g1[2] = (int)((unsigned)rows << 16);                      // tensor_dim1[15:0]
  g1[3] = (int)((unsigned)cols << 16);                      // tile_dim0
  g1[4] = (int)(unsigned)rows;                              // tile_dim1 (tile_dim2=0)
  g1[5] = (int)cols;                                        // tensor_dim0_stride[31:0]
  g1[6] = 0;
  g1[7] = 0;
  const i32x4 z4 = {0, 0, 0, 0};
  const i32x8 z8 = {0, 0, 0, 0, 0, 0, 0, 0};
  __builtin_amdgcn_tensor_load_to_lds(g0, g1, z4, z4, z8, 0);
}

// ---------------------------------------------------------------------------
// 4) GEMM via V_WMMA_F32_16X16X4_F32 (fp32: problem is BW-bound, so keep full
//    precision). One wave computes a 16-row x O strip: each A element is read
//    exactly once (min HBM traffic), W lives in LDS shared by all 8 waves.
//    W staged via TDM when no zero-pad is needed (CW==CPAD), else cooperative
//    padded fill.  A 16x4 layout: lane half selects the K pair; C/D layout:
//    VGPR r -> row r + 8*half, col = lane%16 (ISA 7.12.2).
// ---------------------------------------------------------------------------
template <int CPAD, int CW, int O>
__global__ void __launch_bounds__(256) wmma_mm_kernel(const float* __restrict__ x,
                                                      const float* __restrict__ w,
                                                      const float* __restrict__ bias,
                                                      float* __restrict__ y) {
  constexpr int NT = O / 16;
  __shared__ float wlds[O * CPAD];
  const int tid = threadIdx.x;

  if constexpr (CW == CPAD) {
    if (tid < 32) {                       // wave 0 issues the TDM transfer
      tdm_load_2d(w, wlds, O, CW);
      __builtin_amdgcn_s_wait_tensorcnt(0);
    }
  } else {
    for (int idx = tid; idx < O * CPAD; idx += 256) {
      const int o = idx / CPAD, c = idx % CPAD;
      wlds[idx] = (c < CW) ? w[o * CW + c] : 0.0f;
    }
  }
  __syncthreads();

  const int lane = tid & 31, wv = tid >> 5;
  const int half = lane >> 4;             // 0: K pair {k,k+1}, 1: {k+2,k+3}
  const int lr   = lane & 15;
  const size_t mbase = ((size_t)blockIdx.x * 8 + wv) * 16;
  const float* xrow = x + (mbase + lr) * CPAD + half * 2;

  v8f acc[NT];
#pragma unroll
  for (int nt = 0; nt < NT; ++nt) acc[nt] = {};

#pragma unroll
  for (int k0 = 0; k0 < CPAD; k0 += 4) {
    const v2f a = *(const v2f*)(xrow + k0);               // A fragment (global)
#pragma unroll
    for (int nt = 0; nt < NT; ++nt) {                     // B fragment (LDS)
      const v2f bb = *(const v2f*)(&wlds[(nt * 16 + lr) * CPAD + k0 + half * 2]);
      acc[nt] = __builtin_amdgcn_wmma_f32_16x16x4_f32(false, a, false, bb,
                                                      (short)0, acc[nt], false, false);
    }
  }

#pragma unroll
  for (int nt = 0; nt < NT; ++nt) {
    const float bv = bias[nt * 16 + lr];
    float* yb = y + (mbase + (size_t)half * 8) * O + nt * 16 + lr;
#pragma unroll
    for (int r = 0; r < 8; ++r) yb[(size_t)r * O] = acc[nt][r] + bv;
  }
}

// ---------------------------------------------------------------------------
// 5) BN stats per channel (deterministic block reduction, no float atomics)
// ---------------------------------------------------------------------------
__global__ void __launch_bounds__(256) bn_stats_kernel(const float* __restrict__ y,
                                                       const float* __restrict__ gamma,
                                                       const float* __restrict__ beta,
                                                       float* __restrict__ ss, int O) {
  const int o = blockIdx.x, tid = threadIdx.x;
  float s = 0.0f, sq = 0.0f;
  for (size_t m = tid; m < M_ROWS; m += 256) {
    const float v = y[m * O + o];
    s += v; sq += v * v;
  }
  __shared__ float sh[256], shq[256];
  sh[tid] = s; shq[tid] = sq;
  __syncthreads();
  for (int st = 128; st > 0; st >>= 1) {
    if (tid < st) { sh[tid] += sh[tid + st]; shq[tid] += shq[tid + st]; }
    __syncthreads();
  }
  if (tid == 0) {
    const float invM = 1.0f / (float)M_ROWS;
    const float mean = sh[0] * invM;
    const float var  = shq[0] * invM - mean * mean;
    const float sc   = gamma[o] * rsqrtf(var + 1e-5f);
    ss[o * 2 + 0] = sc;
    ss[o * 2 + 1] = beta[o] - mean * sc;
  }
}

__global__ void __launch_bounds__(256) bn_apply_relu_kernel(float* __restrict__ y,
                                                            const float* __restrict__ ss,
                                                            size_t total, int O) {
  const size_t i = (size_t)blockIdx.x * blockDim.x + threadIdx.x;
  if (i >= total) return;
  const int o = (int)(i % O);
  const float v = y[i] * ss[o * 2] + ss[o * 2 + 1];
  y[i] = v > 0.0f ? v : 0.0f;
}

// ---------------------------------------------------------------------------
// 6) Max over the 32 samples -> new_points
// ---------------------------------------------------------------------------
__global__ void __launch_bounds__(256) maxpool_kernel(const float* __restrict__ y3,
                                                      float* __restrict__ out) {
  const size_t i = (size_t)blockIdx.x * blockDim.x + threadIdx.x;
  if (i >= (size_t)BATCH * S_OUT * 128) return;
  const int o = (int)(i & 127);
  const size_t bs = i >> 7;
  const float* p = y3 + (bs * KSAMP) * 128 + o;
  float m = p[0];
#pragma unroll
  for (int k = 1; k < KSAMP; ++k) m = fmaxf(m, p[(size_t)k * 128]);
  out[(size_t)BATCH * S_OUT * 3 + i] = m;
}

// ---------------------------------------------------------------------------
extern "C" void kernel_launch(void* const* d_in, const int* in_sizes, int n_in,
                              void* d_out, int out_size, void* d_ws, size_t ws_size,
                              hipStream_t stream) {
  const float* xyz = (const float*)d_in[0];
  const float* pts = (const float*)d_in[1];
  const float* w0 = (const float*)d_in[2];  const float* b0 = (const float*)d_in[3];
  const float* g0 = (const float*)d_in[4];  const float* be0 = (const float*)d_in[5];
  const float* w1 = (const float*)d_in[6];  const float* b1 = (const float*)d_in[7];
  const float* g1 = (const float*)d_in[8];  const float* be1 = (const float*)d_in[9];
  const float* w2 = (const float*)d_in[10]; const float* b2 = (const float*)d_in[11];
  const float* g2 = (const float*)d_in[12]; const float* be2 = (const float*)d_in[13];
  float* out = (float*)d_out;

  char* ws = (char*)d_ws;
  int*   idxbuf = (int*)ws;                                   //  2 MB
  float* x0 = (float*)(ws + (size_t)2097152);                 // 16 MB  [M][8]
  float* y1 = (float*)(ws + (size_t)18874368);                // 134 MB [M][64]
  float* y2 = (float*)(ws + (size_t)153092096);               // 134 MB [M][64]
  float* y3 = (float*)(ws + (size_t)287309824);               // 268 MB [M][128]
  float* ss0 = (float*)(ws + (size_t)555745280);
  float* ss1 = ss0 + 256;
  float* ss2 = ss1 + 256;

  // 1) FPS -> new_xyz written straight into d_out[0 : 16*1024*3]
  fps_kernel<<<BATCH, 256, 0, stream>>>(xyz, out);
  // 2) ball query (one wave per centroid: 16384 waves)
  ballq_kernel<<<(BATCH * S_OUT * 32) / 256, 256, 0, stream>>>(xyz, out, idxbuf);
  // 3) gather/concat
  group_kernel<<<(int)(M_ROWS / 256), 256, 0, stream>>>(xyz, pts, out, idxbuf, x0);

  const int gemm_blocks = (int)(M_ROWS / 128);                // 8 waves x 16 rows each
  // 4) layer 1: 6(->8 pad) -> 64   (padded cooperative W staging)
  wmma_mm_kernel<8, 6, 64><<<gemm_blocks, 256, 0, stream>>>(x0, w0, b0, y1);
  bn_stats_kernel<<<64, 256, 0, stream>>>(y1, g0, be0, ss0, 64);
  bn_apply_relu_kernel<<<(int)(M_ROWS * 64 / 256), 256, 0, stream>>>(y1, ss0, M_ROWS * 64, 64);
  // 5) layer 2: 64 -> 64           (W via Tensor Data Mover)
  wmma_mm_kernel<64, 64, 64><<<gemm_blocks, 256, 0, stream>>>(y1, w1, b1, y2);
  bn_stats_kernel<<<64, 256, 0, stream>>>(y2, g1, be1, ss1, 64);
  bn_apply_relu_kernel<<<(int)(M_ROWS * 64 / 256), 256, 0, stream>>>(y2, ss1, M_ROWS * 64, 64);
  // 6) layer 3: 64 -> 128          (W via Tensor Data Mover)
  wmma_mm_kernel<64, 64, 128><<<gemm_blocks, 256, 0, stream>>>(y2, w2, b2, y3);
  bn_stats_kernel<<<128, 256, 0, stream>>>(y3, g2, be2, ss2, 128);
  bn_apply_relu_kernel<<<(int)(M_ROWS * 128 / 256), 256, 0, stream>>>(y3, ss2, M_ROWS * 128, 128);
  // 7) max over K
  maxpool_kernel<<<(BATCH * S_OUT * 128) / 256, 256, 0, stream>>>(y3, out);
}